// KMeansPP_2791728742437
// MI455X (gfx1250) — compile-verified
//
#include <hip/hip_runtime.h>
#include <math.h>

typedef float v2f __attribute__((ext_vector_type(2)));
typedef float v8f __attribute__((ext_vector_type(8)));

#define KC 64        // clusters
#define DF 128       // feature dim
#define MAXIT 10
#define TOLER 1e-4f
#define LDSTRIDE 132 // 64-bank-conflict-free A-fragment reads
#define RED_BLOCKS 256
#define ACC_BLOCKS 256

__device__ __forceinline__ float waveReduceSum(float a) {
#pragma unroll
  for (int off = 16; off > 0; off >>= 1) a += __shfl_xor(a, off, 32);
  return a;
}

// ---------------------------------------------------------------------------
// WMMA distance kernel. MODE 0: labels[n] = argmin_k (c2[k] - 2 C_k.x_n)
//                       MODE 1: mind2[n]  = x2[n] + min over non-empty k
// Empty-cluster masking is folded into the staged c2 values (+inf), so the
// argmin epilogue is branch-free.
// ---------------------------------------------------------------------------
template <int MODE>
__global__ __launch_bounds__(256) void kDist(
    const float* __restrict__ Cg, const float* __restrict__ c2g,
    const int* __restrict__ emptyg, const float* __restrict__ X,
    const float* __restrict__ x2, int* __restrict__ labels,
    float* __restrict__ mind2, int ntiles) {
  __shared__ float ldsC[KC * LDSTRIDE];
  __shared__ float ldsC2[KC];
  const int tid = threadIdx.x;
  for (int idx = tid; idx < KC * DF; idx += 256)
    ldsC[(idx >> 7) * LDSTRIDE + (idx & 127)] = Cg[idx];
  if (tid < KC) {
    float v = c2g[tid];
    if (MODE == 1 && emptyg[tid]) v = __builtin_inff();
    ldsC2[tid] = v;
  }
  __syncthreads();

  const int wid = tid >> 5, lane = tid & 31;
  const int tile = blockIdx.x * 8 + wid;
  if (tile >= ntiles) return;           // wave-uniform: EXEC stays all-1s
  const int nB = tile * 16 + (lane & 15);
  const int ksel = (lane >> 4) << 1;    // K sub-select per ISA A/B layout

  const v8f z = {0.f, 0.f, 0.f, 0.f, 0.f, 0.f, 0.f, 0.f};
  v8f acc[4];
  acc[0] = z; acc[1] = z; acc[2] = z; acc[3] = z;

  const float* xrow = X + nB * DF;
#pragma unroll 4
  for (int k0 = 0; k0 < DF; k0 += 4) {
    const int kk = k0 + ksel;
    v2f b;
    b.x = xrow[kk];
    b.y = xrow[kk + 1];
#pragma unroll
    for (int mt = 0; mt < 4; ++mt) {
      const float* ap = &ldsC[(mt * 16 + (lane & 15)) * LDSTRIDE + kk];
      v2f a;
      a.x = ap[0];
      a.y = ap[1];
      acc[mt] = __builtin_amdgcn_wmma_f32_16x16x4_f32(
          false, a, false, b, (short)0, acc[mt], false, false);
    }
  }

  float best = __builtin_inff();
  int bidx = 0;
  const int moff = (lane >> 4) << 3;
#pragma unroll
  for (int mt = 0; mt < 4; ++mt) {
#pragma unroll
    for (int r = 0; r < 8; ++r) {
      const int m = mt * 16 + moff + r;
      float s = ldsC2[m] - 2.0f * acc[mt][r];
      if (s < best) { best = s; bidx = m; }
    }
  }
  float ov = __shfl_xor(best, 16, 32);
  int oi = __shfl_xor(bidx, 16, 32);
  if (ov < best || (ov == best && oi < bidx)) { best = ov; bidx = oi; }
  if (lane < 16) {
    if (MODE == 0) labels[nB] = bidx;
    else mind2[nB] = best + x2[nB];
  }
}

// ---------------------------------------------------------------------------
__global__ __launch_bounds__(256) void kRowNorm(const float* __restrict__ M,
                                                float* __restrict__ out,
                                                int nrows) {
  const int wid = threadIdx.x >> 5, lane = threadIdx.x & 31;
  const int row = blockIdx.x * 8 + wid;
  if (row >= nrows) return;
  const float* r = M + row * DF;
  float a = 0.f;
#pragma unroll
  for (int d = lane; d < DF; d += 32) { float v = r[d]; a += v * v; }
  a = waveReduceSum(a);
  if (lane == 0) out[row] = a;
}

__global__ __launch_bounds__(256) void kUpdateMinD(
    const float* __restrict__ X, const float* __restrict__ x2,
    const float* __restrict__ Cj, const float* __restrict__ c2j,
    float* __restrict__ min_d, int n) {
  const int wid = threadIdx.x >> 5, lane = threadIdx.x & 31;
  const int p = blockIdx.x * 8 + wid;
  if (p >= n) return;
  const float* xr = X + p * DF;
  float a = 0.f;
#pragma unroll
  for (int d = lane; d < DF; d += 32) a += xr[d] * Cj[d];
  a = waveReduceSum(a);
  if (lane == 0) {
    float d2 = x2[p] + c2j[0] - 2.0f * a;
    min_d[p] = fminf(min_d[p], d2);
  }
}

__global__ __launch_bounds__(128) void kSetCentroid(
    const float* __restrict__ X, float* __restrict__ C, int ci,
    const int* __restrict__ farPtr, int immIdx, float* __restrict__ c2j) {
  __shared__ float red[DF];
  const int idx = farPtr ? *farPtr : immIdx;
  const int t = threadIdx.x;
  float v = X[idx * DF + t];
  C[ci * DF + t] = v;
  red[t] = v * v;
  __syncthreads();
  for (int off = 64; off > 0; off >>= 1) {
    if (t < off) red[t] += red[t + off];
    __syncthreads();
  }
  if (t == 0) c2j[0] = red[0];
}

__global__ __launch_bounds__(256) void kArgmaxPartial(
    const float* __restrict__ vals, int n, float* __restrict__ pVal,
    int* __restrict__ pIdx) {
  __shared__ float sv[256];
  __shared__ int si[256];
  const int t = threadIdx.x;
  float bv = -__builtin_inff();
  int bi = 0x7fffffff;
  for (int i = blockIdx.x * 256 + t; i < n; i += gridDim.x * 256) {
    float v = vals[i];
    if (v > bv) { bv = v; bi = i; }   // ascending i -> first occurrence kept
  }
  sv[t] = bv; si[t] = bi;
  __syncthreads();
  for (int off = 128; off > 0; off >>= 1) {
    if (t < off) {
      if (sv[t + off] > sv[t] || (sv[t + off] == sv[t] && si[t + off] < si[t])) {
        sv[t] = sv[t + off]; si[t] = si[t + off];
      }
    }
    __syncthreads();
  }
  if (t == 0) { pVal[blockIdx.x] = sv[0]; pIdx[blockIdx.x] = si[0]; }
}

__global__ __launch_bounds__(256) void kArgmaxFinal(
    const float* __restrict__ pVal, const int* __restrict__ pIdx, int nblocks,
    int* __restrict__ farIdx) {
  __shared__ float sv[256];
  __shared__ int si[256];
  const int t = threadIdx.x;
  if (t < nblocks) { sv[t] = pVal[t]; si[t] = pIdx[t]; }
  else { sv[t] = -__builtin_inff(); si[t] = 0x7fffffff; }
  __syncthreads();
  for (int off = 128; off > 0; off >>= 1) {
    if (t < off) {
      if (sv[t + off] > sv[t] || (sv[t + off] == sv[t] && si[t + off] < si[t])) {
        sv[t] = sv[t + off]; si[t] = si[t + off];
      }
    }
    __syncthreads();
  }
  if (t == 0) *farIdx = si[0];
}

__global__ void kZeroF(float* __restrict__ p, int n) {
  int i = blockIdx.x * blockDim.x + threadIdx.x;
  if (i < n) p[i] = 0.f;
}
__global__ void kFillF(float* __restrict__ p, int n, float v) {
  int i = blockIdx.x * blockDim.x + threadIdx.x;
  if (i < n) p[i] = v;
}
__global__ void kInitDone(int* __restrict__ done) { *done = 0; }

// LDS-accumulated segment sum, flushed with global f32 atomics.
__global__ __launch_bounds__(256) void kAccum(
    const float* __restrict__ X, const int* __restrict__ labels,
    float* __restrict__ sums, float* __restrict__ counts, int n, int chunk) {
  __shared__ float sS[KC * DF];
  __shared__ float sC[KC];
  const int t = threadIdx.x;
  for (int i = t; i < KC * DF; i += 256) sS[i] = 0.f;
  if (t < KC) sC[t] = 0.f;
  __syncthreads();
  const int half = t >> 7;
  const int d = t & 127;
  const int p0 = blockIdx.x * chunk;
  int pend = p0 + chunk;
  if (pend > n) pend = n;
  for (int p = p0 + half; p < pend; p += 2) {
    int lbl = labels[p];
    atomicAdd(&sS[lbl * DF + d], X[p * DF + d]);
    if (d == 0) atomicAdd(&sC[lbl], 1.0f);
  }
  __syncthreads();
  for (int i = t; i < KC * DF; i += 256) {
    float v = sS[i];
    if (v != 0.f) atomicAdd(&sums[i], v);
  }
  if (t < KC) {
    float v = sC[t];
    if (v != 0.f) atomicAdd(&counts[t], v);
  }
}

__global__ __launch_bounds__(256) void kMeans(
    const float* __restrict__ sums, const float* __restrict__ counts,
    const float* __restrict__ C, float* __restrict__ Cnew,
    int* __restrict__ empty, float* __restrict__ shiftOut,
    int* __restrict__ anyEmptyOut) {
  __shared__ float red[256];
  __shared__ int ae;
  const int t = threadIdx.x;
  if (t == 0) ae = 0;
  __syncthreads();
  float ls = 0.f;
  for (int i = t; i < KC * DF; i += 256) {
    int k = i >> 7;
    float cnt = counts[k];
    float mean = sums[i] / fmaxf(cnt, 1.0f);
    Cnew[i] = mean;
    if (cnt != 0.f) { float dd = mean - C[i]; ls += dd * dd; }
  }
  if (t < KC) {
    int e = (counts[t] == 0.f) ? 1 : 0;
    empty[t] = e;
    if (e) ae = 1;
  }
  red[t] = ls;
  __syncthreads();
  for (int off = 128; off > 0; off >>= 1) {
    if (t < off) red[t] += red[t + off];
    __syncthreads();
  }
  if (t == 0) { shiftOut[0] = red[0]; anyEmptyOut[0] = ae; }
}

__global__ __launch_bounds__(256) void kFinalize(
    const float* __restrict__ X, const float* __restrict__ Cnew,
    const int* __restrict__ empty, const int* __restrict__ farIdx,
    const float* __restrict__ shift, const int* __restrict__ anyEmpty,
    float* __restrict__ C, int* __restrict__ done) {
  const int t = threadIdx.x;
  const int doneOld = *done;
  const int far = *farIdx;
  for (int i = t; i < KC * DF; i += 256) {
    int k = i >> 7, d = i & 127;
    float v = Cnew[i];
    if (empty[k]) v = X[far * DF + d];
    if (!doneOld) C[i] = v;
  }
  __syncthreads();
  if (t == 0) {
    int conv = (shift[0] <= TOLER) && (anyEmpty[0] == 0);
    *done = doneOld | conv;
  }
}

// ---------------------------------------------------------------------------
extern "C" void kernel_launch(void* const* d_in, const int* in_sizes, int n_in,
                              void* d_out, int out_size, void* d_ws,
                              size_t ws_size, hipStream_t stream) {
  (void)n_in; (void)out_size; (void)ws_size;
  const float* X = (const float*)d_in[0];
  const int n = in_sizes[0] / DF;  // 200000
  float* C = (float*)d_out;        // centroids live directly in d_out

  char* ws = (char*)d_ws;
  size_t off = 0;
  auto alloc = [&](size_t bytes) {
    void* p = ws + off;
    off = (off + bytes + 255) & ~(size_t)255;
    return p;
  };
  float* x2     = (float*)alloc((size_t)n * 4);
  float* min_d  = (float*)alloc((size_t)n * 4);
  float* mind2  = (float*)alloc((size_t)n * 4);
  int*   labels = (int*)alloc((size_t)n * 4);
  float* pVal   = (float*)alloc(RED_BLOCKS * 4);
  int*   pIdx   = (int*)alloc(RED_BLOCKS * 4);
  float* sums   = (float*)alloc(KC * DF * 4);
  float* counts = (float*)alloc(KC * 4);
  float* Cnew   = (float*)alloc(KC * DF * 4);
  float* c2     = (float*)alloc(KC * 4);
  float* c2b    = (float*)alloc(KC * 4);
  int*   empty  = (int*)alloc(KC * 4);
  float* c2j    = (float*)alloc(4);
  int*   farIdx = (int*)alloc(4);
  float* shift  = (float*)alloc(4);
  int*   anyE   = (int*)alloc(4);
  int*   done   = (int*)alloc(4);

  const int ntiles = n / 16;
  const int distBlocks = (ntiles + 7) / 8;
  const int ptWaveBlocks = (n + 7) / 8;
  const int elemBlocks = (n + 255) / 256;
  const int chunk = (n + ACC_BLOCKS - 1) / ACC_BLOCKS;

  kInitDone<<<1, 1, 0, stream>>>(done);
  kRowNorm<<<ptWaveBlocks, 256, 0, stream>>>(X, x2, n);
  kFillF<<<elemBlocks, 256, 0, stream>>>(min_d, n, (float)INFINITY);

  // --- farthest-point init (incremental min-distance == reference's masked min)
  kSetCentroid<<<1, 128, 0, stream>>>(X, C, 0, nullptr, 0, c2j);
  kUpdateMinD<<<ptWaveBlocks, 256, 0, stream>>>(X, x2, C, c2j, min_d, n);
  for (int i = 1; i < KC; ++i) {
    kArgmaxPartial<<<RED_BLOCKS, 256, 0, stream>>>(min_d, n, pVal, pIdx);
    kArgmaxFinal<<<1, 256, 0, stream>>>(pVal, pIdx, RED_BLOCKS, farIdx);
    kSetCentroid<<<1, 128, 0, stream>>>(X, C, i, farIdx, 0, c2j);
    kUpdateMinD<<<ptWaveBlocks, 256, 0, stream>>>(X, x2, C + i * DF, c2j,
                                                  min_d, n);
  }

  // --- Lloyd iterations (all launched; device-side `done` flag freezes C)
  for (int t = 0; t < MAXIT; ++t) {
    kRowNorm<<<8, 256, 0, stream>>>(C, c2, KC);
    kDist<0><<<distBlocks, 256, 0, stream>>>(C, c2, empty, X, x2, labels,
                                             mind2, ntiles);
    kZeroF<<<(KC * DF + 255) / 256, 256, 0, stream>>>(sums, KC * DF);
    kZeroF<<<1, 256, 0, stream>>>(counts, KC);
    kAccum<<<ACC_BLOCKS, 256, 0, stream>>>(X, labels, sums, counts, n, chunk);
    kMeans<<<1, 256, 0, stream>>>(sums, counts, C, Cnew, empty, shift, anyE);
    kRowNorm<<<8, 256, 0, stream>>>(Cnew, c2b, KC);
    kDist<1><<<distBlocks, 256, 0, stream>>>(Cnew, c2b, empty, X, x2, labels,
                                             mind2, ntiles);
    kArgmaxPartial<<<RED_BLOCKS, 256, 0, stream>>>(mind2, n, pVal, pIdx);
    kArgmaxFinal<<<1, 256, 0, stream>>>(pVal, pIdx, RED_BLOCKS, farIdx);
    kFinalize<<<1, 256, 0, stream>>>(X, Cnew, empty, farIdx, shift, anyE, C,
                                     done);
  }
}